// PlanLoss_39041252721029
// MI455X (gfx1250) — compile-verified
//
#include <hip/hip_runtime.h>
#include <math.h>

#define EPSF 1e-7f
#define FOUR_OVER_PI2 0.405284734569351f  // 4 / pi^2

typedef __attribute__((ext_vector_type(2))) float v2f;
typedef __attribute__((ext_vector_type(8))) float v8f;

// ---------------------------------------------------------------------------
// Exact 32-lane (wave32) sum using V_WMMA_F32_16X16X4_F32.
// A-matrix layout (32-bit, 16x4): VGPR0 lanes 0-15 -> A[m][0], lanes 16-31 ->
// A[m][2]; VGPR1 -> K=1/K=3 (zeroed). With B = all-ones (4x16),
// D[m][n] = A[m][0] + A[m][2] = p_m + p_{m+16} for every column n.
// D layout: lane n(0-15) VGPR j = rowsum_j ; lane n(16-31) VGPR j = rowsum_{8+j}.
// Sum the 8 D VGPRs per lane, then add across the two lane halves.
// Must be executed with EXEC = all ones (call outside divergent control flow).
// ---------------------------------------------------------------------------
__device__ __forceinline__ float wave_sum32(float p) {
#if __has_builtin(__builtin_amdgcn_wmma_f32_16x16x4_f32)
    v2f a; a.x = p;    a.y = 0.0f;
    v2f b; b.x = 1.0f; b.y = 1.0f;
    v8f c = {};
    v8f d = __builtin_amdgcn_wmma_f32_16x16x4_f32(
        /*neg_a=*/false, a, /*neg_b=*/false, b,
        /*c_mod=*/(short)0, c, /*reuse_a=*/false, /*reuse_b=*/false);
    float s = ((d[0] + d[1]) + (d[2] + d[3])) + ((d[4] + d[5]) + (d[6] + d[7]));
    s += __shfl_xor(s, 16);   // combine rows 0-7 half with rows 8-15 half
    return s;
#else
    float s = p;
    for (int m = 16; m > 0; m >>= 1) s += __shfl_xor(s, m);
    return s;
#endif
}

// Block (256 threads = 8 wave32) sum; result valid on threadIdx.x == 0.
__device__ __forceinline__ float block_sum256(float p, float* lds8) {
    float ws = wave_sum32(p);
    const int wave = threadIdx.x >> 5;
    const int lane = threadIdx.x & 31;
    if (lane == 0) lds8[wave] = ws;
    __syncthreads();
    float tot = 0.0f;
    if (threadIdx.x == 0) {
        #pragma unroll
        for (int i = 0; i < 8; ++i) tot += lds8[i];
    }
    return tot;
}

// ---------------------------------------------------------------------------
// Workspace layout (floats): ws[0]=sum(dist*valid) ws[1]=sum(valid) ws[2]=col acc
// ---------------------------------------------------------------------------
__global__ void zero_ws_kernel(float* ws) {
    if (threadIdx.x < 3) ws[threadIdx.x] = 0.0f;
}

// ------------------------- dist_loss partials ------------------------------
__global__ __launch_bounds__(256) void dist_kernel(const float* __restrict__ pred,
                                                   const float* __restrict__ tgt,
                                                   float* __restrict__ ws, int NT) {
    __shared__ float l0[8];
    __shared__ float l1[8];
    const int idx = blockIdx.x * 256 + threadIdx.x;
    float dsum = 0.0f, csum = 0.0f;
    if (idx < NT) {
        const float2 t2 = ((const float2*)tgt)[idx];
        const float2 p2 = ((const float2*)pred)[idx];
        if (t2.x != -999.0f && t2.y != -999.0f) {
            const float dx = p2.x - t2.x, dy = p2.y - t2.y;
            dsum = sqrtf(dx * dx + dy * dy);
            csum = 1.0f;
        }
    }
    // WMMA reductions: outside divergent region -> full EXEC in every wave.
    const float bd = block_sum256(dsum, l0);
    const float bc = block_sum256(csum, l1);
    if (threadIdx.x == 0) {
        atomicAdd(&ws[0], bd);
        atomicAdd(&ws[1], bc);
    }
}

// ------------------------- col_loss: one block per n -----------------------
#define MAXT 128
#define MAXA 128

// Weighted 3-variant CIoU contribution for one (agent point, timestep).
// Ego geometry and per-(a,w) padded boxes + arctan terms come from LDS.
__device__ __forceinline__ float ciou3_point(float ax, float ay, int a, int t,
                                             const float* __restrict__ exS,
                                             const float* __restrict__ eyS,
                                             const float* __restrict__ awS,
                                             const float* __restrict__ ahS,
                                             const float* __restrict__ vS,
                                             float ehw, float ehh, float egoArea) {
    const float wts[3] = {1.0f, 0.4f, 0.1f};
    const float exv = exS[t], eyv = eyS[t];
    const float dx = ax - exv, dy = ay - eyv;
    const float rho2 = dx * dx + dy * dy;
    const float b1x1 = exv - ehw, b1x2 = exv + ehw;
    const float b1y1 = eyv - ehh, b1y2 = eyv + ehh;
    float acc = 0.0f;
    #pragma unroll
    for (int w = 0; w < 3; ++w) {
        const int j = a * 3 + w;
        const float aw = awS[j], ah = ahS[j], v = vS[j];
        const float hw = 0.5f * aw, hh = 0.5f * ah;
        const float b2x1 = ax - hw, b2x2 = ax + hw;
        const float b2y1 = ay - hh, b2y2 = ay + hh;
        const float iw = fmaxf(fminf(b1x2, b2x2) - fmaxf(b1x1, b2x1), 0.0f);
        const float ih = fmaxf(fminf(b1y2, b2y2) - fmaxf(b1y1, b2y1), 0.0f);
        const float inter = iw * ih;
        const float uni   = egoArea + aw * ah - inter + EPSF;
        const float iou   = inter / uni;
        const float cw = fmaxf(b1x2, b2x2) - fminf(b1x1, b2x1);
        const float ch = fmaxf(b1y2, b2y2) - fminf(b1y1, b2y1);
        const float c2 = cw * cw + ch * ch + EPSF;
        const float alpha = v / (1.0f - iou + v + EPSF);
        acc += wts[w] * (iou - rho2 / c2 - alpha * v);
    }
    return acc;
}

__global__ __launch_bounds__(256) void col_kernel(const float* __restrict__ pred_traj,
                                                  const float* __restrict__ tgt_traj,
                                                  const float* __restrict__ ego_box,
                                                  const float* __restrict__ agent_box,
                                                  const float* __restrict__ agent_traj,
                                                  float* __restrict__ ws,
                                                  int N, int A, int T) {
    __shared__ float exS[MAXT], eyS[MAXT], vaS[MAXT];
    __shared__ float awS[MAXA * 3], ahS[MAXA * 3], vS[MAXA * 3];
    __shared__ float l8[8];

    const int n   = blockIdx.x;
    const int tid = threadIdx.x;

    const float ew = ego_box[2 * n];
    const float eh = ego_box[2 * n + 1];
    const float atanE = atanf(ew / (eh + EPSF));   // hoisted: per-n only

    const float tld[3] = {0.0f, 0.0048828125f, 0.009765625f};

    // Stage ego trajectory + validity mask into LDS (coalesced float2 loads).
    for (int t = tid; t < T; t += 256) {
        const size_t base = ((size_t)n * T + t);
        const float2 e2 = ((const float2*)pred_traj)[base];
        const float2 g2 = ((const float2*)tgt_traj)[base];
        exS[t] = e2.x;
        eyS[t] = e2.y;
        vaS[t] = (g2.x != -999.0f && g2.y != -999.0f) ? 1.0f : 0.0f;
    }
    // Precompute per-(a,w): padded box sizes and the arctan aspect term v.
    // Hoists 12.6M atans down to A*3 per block.
    for (int i = tid; i < A * 3; i += 256) {
        const int a = i / 3, w = i - 3 * a;
        const float2 b2 = ((const float2*)agent_box)[(size_t)n * A + a];
        const float aw = b2.x + tld[w];
        const float ah = b2.y + tld[w];
        awS[i] = aw;
        ahS[i] = ah;
        const float dAt = atanf(aw / (ah + EPSF)) - atanE;
        vS[i] = FOUR_OVER_PI2 * dAt * dAt;
    }
    __syncthreads();

    const float ehw = 0.5f * ew, ehh = 0.5f * eh;
    const float egoArea = ew * eh;

    float acc = 0.0f;
    if ((T & 1) == 0) {
        // Fast path: B128 streaming — each lane reads two consecutive
        // timesteps (16 B). Lane-consecutive pairs => fully coalesced; pairs
        // never straddle an agent row since T is even.
        const int T2 = T >> 1;
        const float4* tr4 = (const float4*)agent_traj;
        const int stride16 = 256;  // elements of float4 per block step
        for (int i = tid; i < A * T2; i += 256) {
            const int a = i / T2, tp = i - a * T2;
            const int t0 = tp * 2;
            const size_t qidx = ((size_t)n * A + a) * T2 + tp;
            // Prefetch one block-stride ahead on the dominant HBM stream.
            __builtin_prefetch(&tr4[qidx + stride16], 0, 0);
            const float4 q = tr4[qidx];
            if (vaS[t0] != 0.0f)
                acc += ciou3_point(q.x, q.y, a, t0,     exS, eyS, awS, ahS, vS,
                                   ehw, ehh, egoArea);
            if (vaS[t0 + 1] != 0.0f)
                acc += ciou3_point(q.z, q.w, a, t0 + 1, exS, eyS, awS, ahS, vS,
                                   ehw, ehh, egoArea);
        }
    } else {
        // General path: float2 per point.
        const float2* tr2 = (const float2*)agent_traj;
        for (int i = tid; i < A * T; i += 256) {
            const int a = i / T, t = i - a * T;
            const float2 at = tr2[((size_t)n * A + a) * T + t];
            if (vaS[t] != 0.0f)
                acc += ciou3_point(at.x, at.y, a, t, exS, eyS, awS, ahS, vS,
                                   ehw, ehh, egoArea);
        }
    }

    // WMMA block reduction (full EXEC here).
    const float bs = block_sum256(acc, l8);
    if (tid == 0) {
        float denom = 0.0f;             // per-n only: sum of valid mask over T
        for (int t = 0; t < T; ++t) denom += vaS[t];
        atomicAdd(&ws[2], bs / denom);  // contributes sum_{a,w,t} w*ciou / denom_n
    }
}

// ------------------------- finalize ----------------------------------------
__global__ void finalize_kernel(const float* __restrict__ ws,
                                float* __restrict__ out, float invNA) {
    if (blockIdx.x == 0 && threadIdx.x == 0) {
        const float dist_loss = ws[0] / ws[1];
        const float col_loss  = ws[2] * invNA;   // mean over N*A
        out[0] = 1.0f * dist_loss + 0.5f * col_loss;  // DIST_LAMBDA, COL_LAMBDA
    }
}

extern "C" void kernel_launch(void* const* d_in, const int* in_sizes, int n_in,
                              void* d_out, int out_size, void* d_ws, size_t ws_size,
                              hipStream_t stream) {
    const float* pred_traj  = (const float*)d_in[0];  // (N, T, 2)
    const float* tgt_traj   = (const float*)d_in[1];  // (N, T, 2)
    const float* ego_box    = (const float*)d_in[2];  // (N, 2)
    const float* agent_box  = (const float*)d_in[3];  // (N, A, 2)
    const float* agent_traj = (const float*)d_in[4];  // (N, A, T, 2)

    const int N = in_sizes[2] / 2;
    const int A = in_sizes[3] / (2 * N);
    const int T = in_sizes[0] / (2 * N);

    float* ws = (float*)d_ws;

    zero_ws_kernel<<<1, 32, 0, stream>>>(ws);

    const int NT = N * T;
    dist_kernel<<<(NT + 255) / 256, 256, 0, stream>>>(pred_traj, tgt_traj, ws, NT);

    col_kernel<<<N, 256, 0, stream>>>(pred_traj, tgt_traj, ego_box,
                                      agent_box, agent_traj, ws, N, A, T);

    finalize_kernel<<<1, 32, 0, stream>>>(ws, (float*)d_out,
                                          1.0f / (float)(N * A));
}